// GCN_V2E2V_layer_66022237274402
// MI455X (gfx1250) — compile-verified
//
#include <hip/hip_runtime.h>
#include <hip/hip_bf16.h>

#define FEAT 32
#define SIZE_E_CONST 200000

typedef __attribute__((ext_vector_type(2))) float v2f;
typedef __attribute__((ext_vector_type(8))) float v8f;

// ---------------- zero fill ----------------
__global__ void hg_zero_kernel(float* __restrict__ p, long n) {
    long i = (long)blockIdx.x * blockDim.x + threadIdx.x;
    long stride = (long)gridDim.x * blockDim.x;
    for (; i < n; i += stride) p[i] = 0.0f;
}

// ---------------- degree counting ----------------
__global__ void hg_degree_kernel(const int* __restrict__ edge_v,
                                 const int* __restrict__ edge_e,
                                 float* __restrict__ deg_v,
                                 float* __restrict__ deg_e, int n) {
    int i = blockIdx.x * blockDim.x + threadIdx.x;
    if (i < n) {
        atomicAdd(&deg_v[edge_v[i]], 1.0f);
        atomicAdd(&deg_e[edge_e[i]], 1.0f);
    }
}

// ---------------- gather/scatter: dst[di[inc]][d] += src[si[inc]][d] ----------------
// lane = feature dim -> 32 consecutive f32 atomics = one L2 cacheline per wave
__global__ void hg_scatter_kernel(const int* __restrict__ src_idx,
                                  const int* __restrict__ dst_idx,
                                  const float* __restrict__ src,
                                  float* __restrict__ dst, long total) {
    long t = (long)blockIdx.x * blockDim.x + threadIdx.x;
    if (t >= total) return;
    int inc = (int)(t >> 5);
    int d   = (int)(t & 31);
    int s  = src_idx[inc];   // wave-uniform broadcast load
    int dd = dst_idx[inc];
    float val = src[(long)s * FEAT + d];
    atomicAdd(&dst[(long)dd * FEAT + d], val);
}

// ---------------- x_e = relu(sum / deg_e), in place ----------------
__global__ void hg_scale_relu_kernel(float* __restrict__ xe,
                                     const float* __restrict__ deg_e, long total) {
    long t = (long)blockIdx.x * blockDim.x + threadIdx.x;
    if (t >= total) return;
    int e = (int)(t >> 5);
    float dg = deg_e[e];
    float v = xe[t];
    xe[t] = (dg > 0.0f) ? fmaxf(v / dg, 0.0f) : 0.0f;
}

// ---------------- finalize: scale+relu, row norms via WMMA Gram, normalize ----------------
// One wave per 16-row tile. 4 waves per block (128 threads).
__global__ void hg_finalize_kernel(float* __restrict__ xv,   // in: raw sums (d_out buffer), out: normalized
                                   const float* __restrict__ deg_v,
                                   int size_v) {
    __shared__ float norms2[4][16];

    const int lane = threadIdx.x & 31;
    const int w    = threadIdx.x >> 5;
    const int numTiles = (size_v + 15) >> 4;
    const int tile = blockIdx.x * 4 + w;
    const bool tileValid = (tile < numTiles);
    const int tilec = tileValid ? tile : 0;

    const int r  = lane & 15;        // row within tile
    const int hi = lane >> 4;        // lane half selects K pair {0,1} vs {2,3}
    int row = tilec * 16 + r;
    const bool rowValid = tileValid && (row < size_v);
    int rowL = row < size_v ? row : (size_v - 1);   // clamp for loads, keep EXEC full

    float dg = deg_v[rowL];
    float inv_d = (dg > 0.0f) ? (1.0f / dg) : 0.0f;

    // Load this lane's share of the 16x32 tile: cols {4j+2*hi, 4j+2*hi+1}
    const float* rp = xv + (long)rowL * FEAT;
    v2f a[8];
#pragma unroll
    for (int j = 0; j < 8; ++j) {
        int c = 4 * j + 2 * hi;
        a[j].x = fmaxf(rp[c]     * inv_d, 0.0f);
        a[j].y = fmaxf(rp[c + 1] * inv_d, 0.0f);
    }

    // Gram = X * X^T via 8x V_WMMA_F32_16X16X4_F32 (A and B register layouts coincide
    // for X vs X^T, so pass the same registers). Diagonal = squared row norms.
    v8f acc = {};
#pragma unroll
    for (int j = 0; j < 8; ++j) {
        acc = __builtin_amdgcn_wmma_f32_16x16x4_f32(
            /*neg_a=*/false, a[j], /*neg_b=*/false, a[j],
            /*c_mod=*/(short)0, acc, /*reuse_a=*/false, /*reuse_b=*/false);
    }

    // Diagonal (m,m): m<8 -> VGPR m @ lane m ; m>=8 -> VGPR m-8 @ lane m+16
    if (lane < 8)   norms2[w][lane]      = acc[lane];
    if (lane >= 24) norms2[w][lane - 16] = acc[lane - 24];
    __syncthreads();

    float n2  = norms2[w][r];
    float inv_n = 1.0f / fmaxf(sqrtf(n2), 1e-12f);

    if (rowValid) {
        float* op = xv + (long)row * FEAT;
#pragma unroll
        for (int j = 0; j < 8; ++j) {
            int c = 4 * j + 2 * hi;
            op[c]     = a[j].x * inv_n;
            op[c + 1] = a[j].y * inv_n;
        }
    }
}

extern "C" void kernel_launch(void* const* d_in, const int* in_sizes, int n_in,
                              void* d_out, int out_size, void* d_ws, size_t ws_size,
                              hipStream_t stream) {
    const float* x    = (const float*)d_in[0];
    const int*   edge = (const int*)d_in[1];   // (3, N) int32 row-major
    (void)x;

    const int N      = in_sizes[1] / 3;          // 1.6M incidences
    const int size_v = in_sizes[0] / FEAT;       // 100000
    const int size_e = SIZE_E_CONST;             // 200000 (reference constant)

    const int* edge_v = edge;            // vertex id per incidence
    const int* edge_e = edge + N;        // hyperedge id per incidence

    // Workspace layout (floats): deg_e | deg_v | x_e
    float* ws    = (float*)d_ws;
    float* deg_e = ws;
    float* deg_v = ws + size_e;
    float* x_e   = ws + size_e + size_v;
    float* xv    = (float*)d_out;        // d_out doubles as the x_v accumulator

    const long ws_words  = (long)size_e + size_v + (long)size_e * FEAT;
    const long out_words = (long)size_v * FEAT;

    // 1) zero accumulators (must run every call: harness does not re-poison)
    hg_zero_kernel<<<4096, 256, 0, stream>>>(ws, ws_words);
    hg_zero_kernel<<<4096, 256, 0, stream>>>(xv, out_words);

    // 2) degrees
    hg_degree_kernel<<<(N + 255) / 256, 256, 0, stream>>>(edge_v, edge_e, deg_v, deg_e, N);

    // 3) v2e scatter: x_e[e] += x[v]
    {
        long total = (long)N * FEAT;
        int blocks = (int)((total + 255) / 256);
        hg_scatter_kernel<<<blocks, 256, 0, stream>>>(edge_v, edge_e, (const float*)d_in[0], x_e, total);
    }

    // 4) x_e = relu(x_e / deg_e)
    {
        long total = (long)size_e * FEAT;
        int blocks = (int)((total + 255) / 256);
        hg_scale_relu_kernel<<<blocks, 256, 0, stream>>>(x_e, deg_e, total);
    }

    // 5) e2v scatter: xv[v] += x_e[e]
    {
        long total = (long)N * FEAT;
        int blocks = (int)((total + 255) / 256);
        hg_scatter_kernel<<<blocks, 256, 0, stream>>>(edge_e, edge_v, x_e, xv, total);
    }

    // 6) scale + relu + WMMA row-norm + normalize
    {
        int numTiles = (size_v + 15) / 16;
        int blocks = (numTiles + 3) / 4;
        hg_finalize_kernel<<<blocks, 128, 0, stream>>>(xv, deg_v, size_v);
    }
}